// Spherical_73126113181899
// MI455X (gfx1250) — compile-verified
//
#include <hip/hip_runtime.h>
#include <hip/hip_bf16.h>
#include <math.h>

// out = exp(s) * x via V_WMMA_F32_16X16X4_F32 with an in-register scaled-identity
// A matrix (bit-exact vs fp32 reference: identity chunks add exact zeros, one RNE
// rounding of e*x). Memory-bound: 512MB streamed @ 23.3 TB/s => ~22us floor.
// NT cache hints on both streams (touched once; 512MB > 192MB L2).
// This revision folds all per-lane address math into one load pointer and one
// store pointer so the 8 loads / 8 stores become single s_clause bursts with
// immediate IOFFSETs (kills the per-load v_lshl_add_u64 chains seen in round 1).

typedef __attribute__((ext_vector_type(2))) float v2f;
typedef __attribute__((ext_vector_type(8))) float v8f;

#define D_DIM 1024            // reference feature dim (fixed)
#define TILE 16
#define TCOLS (D_DIM / TILE)  // 64 tiles per 16-row band

__global__ __launch_bounds__(256) void spherical_scale_wmma(
    const float* __restrict__ x,
    const float* __restrict__ spherical,
    float* __restrict__ out,
    int n_tiles)
{
    const float e = __expf(spherical[0]);   // uniform scalar, cached read

    const int lane = threadIdx.x & 31;
    const int m    = lane & 15;   // N column (B/D) or M row (A)
    const int half = lane >> 4;   // 0: lanes 0-15, 1: lanes 16-31

    // A matrix (16x16 scaled identity, split into 4 K-chunks of 4) is
    // tile-invariant: precompute the 8 A register values once.
    // A VGPR j of chunk c holds A[m][k], k = 4c + 2*half + j.
    v2f areg[4];
    #pragma unroll
    for (int c = 0; c < 4; ++c)
        #pragma unroll
        for (int j = 0; j < 2; ++j)
            areg[c][j] = (m == 4 * c + 2 * half + j) ? e : 0.0f;

    const int wavesPerBlock = blockDim.x >> 5;
    int w = blockIdx.x * wavesPerBlock + (int)(threadIdx.x >> 5);
    const int wstride = gridDim.x * wavesPerBlock;

    // w is wave-uniform, so EXEC is all-ones whenever the WMMAs execute.
    for (; w < n_tiles; w += wstride) {
        const int tr = w >> 6;          // tile row band (16 rows)
        const int tc = w & (TCOLS - 1); // tile column
        const size_t base = (size_t)tr * TILE * D_DIM + (size_t)tc * TILE;

        // Fold all per-lane variability into the pointers; remaining offsets
        // are compile-time constants -> single-clause b32 bursts.
        const float* __restrict__ px = x   + base + m + (size_t)(2 * half) * D_DIM;
        float*       __restrict__ po = out + base + m + (size_t)(8 * half) * D_DIM;

        // B chunk c, VGPR j holds x_tile[k][n], k = 4c + 2*half + j, n = m.
        v2f breg[4];
        #pragma unroll
        for (int c = 0; c < 4; ++c)
            #pragma unroll
            for (int j = 0; j < 2; ++j)
                breg[c][j] = __builtin_nontemporal_load(px + (4 * c + j) * D_DIM);

        v8f acc = {};
        #pragma unroll
        for (int c = 0; c < 4; ++c)
            // (neg_a, A, neg_b, B, c_mod, C, reuse_a, reuse_b)
            acc = __builtin_amdgcn_wmma_f32_16x16x4_f32(
                false, areg[c], false, breg[c], (short)0, acc, false, false);

        // D layout: VGPR v -> row v + 8*half, col m.
        #pragma unroll
        for (int v = 0; v < 8; ++v)
            __builtin_nontemporal_store(acc[v], po + v * D_DIM);
    }
}

extern "C" void kernel_launch(void* const* d_in, const int* in_sizes, int n_in,
                              void* d_out, int out_size, void* d_ws, size_t ws_size,
                              hipStream_t stream) {
    const float* x = (const float*)d_in[0];         // [N, 1024] fp32
    const float* s = (const float*)d_in[1];         // [1] fp32
    float* out = (float*)d_out;                     // [N, 1024] fp32

    const long long n_elems = (long long)in_sizes[0];
    const int rows    = (int)(n_elems / D_DIM);     // 65536
    const int n_tiles = (rows / TILE) * TCOLS;      // 262144

    const int threads = 256;                        // 8 wave32s per block
    const int wavesPerBlock = threads / 32;
    int blocks = (n_tiles + wavesPerBlock - 1) / wavesPerBlock;  // 1 tile/wave
    if (blocks < 1) blocks = 1;

    spherical_scale_wmma<<<blocks, threads, 0, stream>>>(x, s, out, n_tiles);
}